// GaussianMixtureLayer_24189255811284
// MI455X (gfx1250) — compile-verified
//
#include <hip/hip_runtime.h>

typedef float v2f __attribute__((ext_vector_type(2)));
typedef float v4f __attribute__((ext_vector_type(4)));
typedef float v8f __attribute__((ext_vector_type(8)));

#define LOG_2PI_F 1.8378770664093453f
#define LOG_K_F   3.4657359027997265f   // ln(32)
#define LOG2E_F   1.4426950408889634f
#define LN2_F     0.6931471805599453f

#define LDS_ROW_STRIDE 36               // floats; conflict-free + 16B aligned
#define WAVES_PER_BLOCK 8

// xor-16 lane swap (partner half-wave), single ds_swizzle_b32:
// group-of-32 bit mode offset = {xor=0x10, or=0, and=0x1f} = 0x401f
__device__ __forceinline__ float swz_xor16(float v) {
    return __int_as_float(__builtin_amdgcn_ds_swizzle(__float_as_int(v), 0x401f));
}
__device__ __forceinline__ float efast(float v, float nml) {
    // exp(v - mx) computed as exp2(v*log2e - mx*log2e): one FMA + one v_exp
    return __builtin_amdgcn_exp2f(__builtin_fmaf(v, LOG2E_F, nml));
}
__device__ __forceinline__ float max4(const v4f t) {
    return fmaxf(fmaxf(t.x, t.y), fmaxf(t.z, t.w));
}

// One wave computes 16-point tiles:
//   comp_lp(16x32) = A(16x4) x [B0|B1](4x32)  via two V_WMMA_F32_16X16X4_F32
// A = [x0, x1, x0^2+x1^2, 1],  B cols = [mu0/s^2, mu1/s^2, -0.5/s^2, c_k]
// => D is comp_lp directly (exact f32 FMA math).  D is then transposed
// through LDS so each lane owns half a row (16 contiguous comps).
__global__ void __launch_bounds__(256)
gmm_wmma_kernel(const float* __restrict__ inputs,   // (N,2)
                const float* __restrict__ mu,       // (32,2)
                const float* __restrict__ sigma,    // (32,1)
                float* __restrict__ out_mix,        // (N,)
                float* __restrict__ out_comp,       // (N,32)
                int N, int tileStride)              // tileStride = total waves
{
    __shared__ float lds[WAVES_PER_BLOCK * 16 * LDS_ROW_STRIDE];

    const int lane = threadIdx.x & 31;
    const int wave = threadIdx.x >> 5;
    const int wgid = blockIdx.x * (blockDim.x >> 5) + wave;

    const bool hi  = lane >= 16;
    const int  l15 = lane & 15;

    float* const wlds = lds + wave * (16 * LDS_ROW_STRIDE);
    // scatter base: this half's first row; r is a +r*144B immediate offset
    float* const sct  = wlds + (hi ? 8 : 0) * LDS_ROW_STRIDE + l15;
    // gather base: lane owns row l15, half `hi` (16 contiguous comps)
    const float* const gat = wlds + l15 * LDS_ROW_STRIDE + (hi ? 16 : 0);

    // ---------------- B matrices (built once per wave) ----------------
    v2f b0, b1;
    {
        int n0 = l15;
        float m00 = mu[2 * n0 + 0], m01 = mu[2 * n0 + 1];
        float s0  = fabsf(sigma[n0]) + 1e-5f;
        float is0 = 1.0f / (s0 * s0);
        if (!hi) { b0.x = m00 * is0;  b0.y = m01 * is0; }
        else {
            b0.x = -0.5f * is0;
            b0.y = -0.5f * (m00 * m00 + m01 * m01) * is0 - 2.0f * __logf(s0) - LOG_2PI_F;
        }
        int n1 = l15 + 16;
        float m10 = mu[2 * n1 + 0], m11 = mu[2 * n1 + 1];
        float s1  = fabsf(sigma[n1]) + 1e-5f;
        float is1 = 1.0f / (s1 * s1);
        if (!hi) { b1.x = m10 * is1;  b1.y = m11 * is1; }
        else {
            b1.x = -0.5f * is1;
            b1.y = -0.5f * (m10 * m10 + m11 * m11) * is1 - 2.0f * __logf(s1) - LOG_2PI_F;
        }
    }

    const long stride = (long)tileStride * 16;

    for (long base = (long)wgid * 16; base < N; base += stride) {
        // ------------- A matrix: 16 points, K padded to 4 -------------
        long p = base + l15;
        if (p >= N) p = (long)N - 1;           // clamp (reads only)
        const v2f xy = *(const v2f*)(inputs + 2 * p);
        v2f a;
        if (!hi) { a.x = xy.x; a.y = xy.y; }
        else     { a.x = xy.x * xy.x + xy.y * xy.y; a.y = 1.0f; }

        v8f c = {};
        v8f d0 = __builtin_amdgcn_wmma_f32_16x16x4_f32(false, a, false, b0, (short)0, c, false, false);
        v8f d1 = __builtin_amdgcn_wmma_f32_16x16x4_f32(false, a, false, b1, (short)0, c, false, false);

        // ---- scatter D to LDS (row-major, stride 36): immediate offsets ----
        #pragma unroll
        for (int r = 0; r < 8; ++r) {
            sct[r * LDS_ROW_STRIDE]      = d0[r];   // comp l15
            sct[r * LDS_ROW_STRIDE + 16] = d1[r];   // comp l15+16
        }

        // ---- gather transposed: 16 contiguous comps of row l15 ----
        const v4f t0 = *(const v4f*)(gat + 0);
        const v4f t1 = *(const v4f*)(gat + 4);
        const v4f t2 = *(const v4f*)(gat + 8);
        const v4f t3 = *(const v4f*)(gat + 12);

        // ---- comp_lp: 4 x global_store_b128, single bounds predicate ----
        const long row = base + l15;
        const bool ok  = row < (long)N;
        float* gp = out_comp + row * 32 + (hi ? 16 : 0);
        if (ok) {
            *(v4f*)(gp + 0)  = t0;
            *(v4f*)(gp + 4)  = t1;
            *(v4f*)(gp + 8)  = t2;
            *(v4f*)(gp + 12) = t3;
        }

        // ---- logsumexp: local tree + one xor-16 combine per phase ----
        float mx = fmaxf(fmaxf(max4(t0), max4(t1)), fmaxf(max4(t2), max4(t3)));
        mx = fmaxf(mx, swz_xor16(mx));
        const float nml = -mx * LOG2E_F;
        float s01 = (efast(t0.x, nml) + efast(t0.y, nml)) +
                    (efast(t0.z, nml) + efast(t0.w, nml)) +
                    (efast(t1.x, nml) + efast(t1.y, nml)) +
                    (efast(t1.z, nml) + efast(t1.w, nml));
        float s23 = (efast(t2.x, nml) + efast(t2.y, nml)) +
                    (efast(t2.z, nml) + efast(t2.w, nml)) +
                    (efast(t3.x, nml) + efast(t3.y, nml)) +
                    (efast(t3.z, nml) + efast(t3.w, nml));
        float sm = s01 + s23;
        sm += swz_xor16(sm);

        const float lp = __builtin_fmaf(__builtin_amdgcn_logf(sm), LN2_F,
                                        mx - LOG_K_F);
        if (ok && !hi)
            out_mix[row] = lp;                  // lanes 0-15: coalesced 64B
    }
}

extern "C" void kernel_launch(void* const* d_in, const int* in_sizes, int n_in,
                              void* d_out, int out_size, void* d_ws, size_t ws_size,
                              hipStream_t stream) {
    const float* inputs = (const float*)d_in[0];  // (N,2) f32
    const float* mu     = (const float*)d_in[1];  // (32,2) f32
    const float* sigma  = (const float*)d_in[2];  // (32,1) f32

    const int N = in_sizes[0] / 2;                // flat count / D
    float* out_mix  = (float*)d_out;              // first N: mix_lp
    float* out_comp = out_mix + N;                // then N*32: comp_lp

    const int nTiles       = (N + 15) / 16;       // 16 points per wave-tile
    const int tilesPerWave = 4;                   // amortize setup
    int blocks = (nTiles + WAVES_PER_BLOCK * tilesPerWave - 1) /
                 (WAVES_PER_BLOCK * tilesPerWave);
    if (blocks < 1) blocks = 1;
    const int totalWaves = blocks * WAVES_PER_BLOCK;

    gmm_wmma_kernel<<<blocks, 256, 0, stream>>>(inputs, mu, sigma,
                                                out_mix, out_comp, N, totalWaves);
}